// SpectralConv2d_fast_LoRA_MoE_40613210751007
// MI455X (gfx1250) — compile-verified
//
#include <hip/hip_runtime.h>
#include <hip/hip_bf16.h>

// ---------------------------------------------------------------------------
// SpectralConv2d + LoRA-MoE for MI455X (gfx1250).
//   - all heavy math on V_WMMA_F32_16X16X4_F32 (fp32 matrix op; complex64
//     reference => fp32 accumulation mandatory for spectral accuracy)
//   - LDS staging via GLOBAL_LOAD_ASYNC_TO_LDS_B128/B32 (ASYNCcnt path)
// ---------------------------------------------------------------------------

#define B_   32
#define CIN_ 64
#define COUT_ 64
#define H_   128
#define W_   128
#define KW_  64      // only frequency columns 0..63 are ever touched
#define M1_  16
#define M2_  16
#define E_   15
#define RANK_ 4

typedef float v2f __attribute__((ext_vector_type(2)));
typedef float v8f __attribute__((ext_vector_type(8)));

static __device__ inline v8f vzero8() {
  v8f z;
#pragma unroll
  for (int j = 0; j < 8; ++j) z[j] = 0.0f;
  return z;
}

// D = A(16x4) * B(4x16) + C, fp32. 8-arg form: (neg_a,A,neg_b,B,c_mod,C,reuse_a,reuse_b)
static __device__ inline v8f wmma4(v2f a, v2f b, v8f c) {
  return __builtin_amdgcn_wmma_f32_16x16x4_f32(false, a, false, b, (short)0, c,
                                               false, false);
}

// A-fragment: tile origin at row 0 / current k column, row-major, stride lda.
// Lane L<16: (m=L, k=0,1); L>=16: (m=L-16, k=2,3).
static __device__ inline v2f load_a(const float* A, int lda) {
  int lane = threadIdx.x & 31;
  int m = lane & 15, half = lane >> 4;
  const float* p = A + m * lda + 2 * half;
  v2f r; r.x = p[0]; r.y = p[1];
  return r;
}

// B-fragment: tile origin at current k row / col 0, row-major, stride ldb.
// VGPR0 = rows {0,1}, VGPR1 = rows {2,3} striped over lane halves.
static __device__ inline v2f load_b(const float* Bm, int ldb) {
  int lane = threadIdx.x & 31;
  int col = lane & 15, half = lane >> 4;
  v2f r; r.x = Bm[half * ldb + col]; r.y = Bm[(half + 2) * ldb + col];
  return r;
}

// C/D fragment scatter: VGPR j holds rows {j, j+8} striped over lane halves.
static __device__ inline void store_c(float* C, int ldc, v8f acc) {
  int lane = threadIdx.x & 31;
  int col = lane & 15, half = lane >> 4;
#pragma unroll
  for (int j = 0; j < 8; ++j) C[(j + 8 * half) * ldc + col] = acc[j];
}

// ---------------------------------------------------------------------------
// Async global->LDS DMA (CDNA5): GLOBAL_LOAD_ASYNC_TO_LDS_*, tracked by
// ASYNCcnt.  GVS addressing: mem = SGPR64 + VGPR32 + ioffset.  The LDS
// destination address is the low 32 bits of the flat pointer (LDS aperture
// keeps the wave-relative LDS offset in addr[31:0]).
// ---------------------------------------------------------------------------
static __device__ inline void async_copy_b128(float* dst_lds,
                                              const float* src, int nfloats) {
  unsigned long long base = (unsigned long long)(const void*)src;
  for (int n = threadIdx.x * 4; n < nfloats; n += 256 * 4) {
    unsigned lds = (unsigned)(unsigned long long)(const void*)(dst_lds + n);
    unsigned off = (unsigned)(n * 4);
    asm volatile("global_load_async_to_lds_b128 %0, %1, %2"
                 :: "v"(lds), "v"(off), "s"(base)
                 : "memory");
  }
}

static __device__ inline void async_ld_b32(float* dst_lds, const float* src,
                                           unsigned off_bytes) {
  unsigned lds = (unsigned)(unsigned long long)(const void*)dst_lds;
  unsigned long long base = (unsigned long long)(const void*)src;
  asm volatile("global_load_async_to_lds_b32 %0, %1, %2"
               :: "v"(lds), "v"(off_bytes), "s"(base)
               : "memory");
}

// wait for this wave's async DMAs, then workgroup barrier
static __device__ inline void async_join() {
  asm volatile("s_wait_asynccnt 0x0" ::: "memory");
  __syncthreads();
}

// ---------------------------------------------------------------------------
// Kernel: build all twiddle tables (one shot, double trig for accuracy).
// ---------------------------------------------------------------------------
__global__ void build_tables(float* CW, float* SW, float* HC, float* HSP,
                             float* HSN, float* WR, float* WI) {
  int idx = blockIdx.x * blockDim.x + threadIdx.x;
  const double TWO_PI = 6.283185307179586476925286766559;
  if (idx < H_ * KW_) {
    int w = idx / KW_, k = idx % KW_;
    double th = TWO_PI * (double)((w * k) & 127) / 128.0;
    CW[idx] = (float)cos(th);
    SW[idx] = (float)(-sin(th));
  }
  if (idx < H_ * H_) {
    int a = idx / H_, h = idx % H_;
    double th = TWO_PI * (double)((a * h) & 127) / 128.0;
    float c = (float)cos(th), s = (float)sin(th);
    HC[idx] = c; HSP[idx] = s; HSN[idx] = -s;
  }
  if (idx < KW_ * W_) {
    int k = idx / W_, w = idx % W_;
    double th = TWO_PI * (double)((k * w) & 127) / 128.0;
    float scale = 1.0f / (float)(H_ * W_);
    float amp = (k == 0) ? scale : 2.0f * scale;
    WR[idx] = amp * (float)cos(th);
    WI[idx] = -amp * (float)sin(th);
  }
}

// ---------------------------------------------------------------------------
// Kernel: transpose dense weights  [i][o][x][y] (complex interleaved)
//   ->  per-frequency-point contiguous split planes  [p=(x*16+y)][i][o]
// ---------------------------------------------------------------------------
__global__ void transpose_w(const float* w, float* tr, float* ti) {
  int idx = blockIdx.x * blockDim.x + threadIdx.x;  // over 64*64*256
  if (idx >= CIN_ * COUT_ * 256) return;
  int i = idx / (COUT_ * 256);
  int rem = idx % (COUT_ * 256);
  int o = rem / 256;
  int p = rem % 256;
  float re = w[2 * idx], im = w[2 * idx + 1];
  int dst = (p * CIN_ + i) * COUT_ + o;
  tr[dst] = re; ti[dst] = im;
}

__global__ void zero_buf(float* p, long n) {
  long i = (long)blockIdx.x * blockDim.x + threadIdx.x;
  long stride = (long)gridDim.x * blockDim.x;
  for (; i < n; i += stride) p[i] = 0.0f;
}

// ---------------------------------------------------------------------------
// Stage 1: per channel ch=b*CIN+i, Yr = X @ CW, Yi = X @ SW
//   X: [128 h][128 w] real,  tables DMA'd into LDS (64 KB).
//   8 waves: wave = M-tile (h), loops 4 N-tiles x 2 planes.
// ---------------------------------------------------------------------------
__global__ __launch_bounds__(256)
void stage_wdft(const float* __restrict__ x, const float* __restrict__ CW,
                const float* __restrict__ SW, float* __restrict__ Yr,
                float* __restrict__ Yi) {
  __shared__ alignas(16) float sB[2 * H_ * KW_];  // CW then SW, 65536 bytes
  long ch = blockIdx.x;
  const float* X = x + ch * (H_ * W_);
  async_copy_b128(sB, CW, H_ * KW_);
  async_copy_b128(sB + H_ * KW_, SW, H_ * KW_);
  async_join();
  int wave = threadIdx.x >> 5;
  const float* A = X + wave * 16 * W_;
  v8f acc[8];
#pragma unroll
  for (int n = 0; n < 8; ++n) acc[n] = vzero8();
  for (int k = 0; k < W_; k += 4) {
    v2f a = load_a(A + k, W_);
#pragma unroll
    for (int nt = 0; nt < 4; ++nt) {
      v2f br = load_b(sB + k * KW_ + nt * 16, KW_);
      v2f bi = load_b(sB + H_ * KW_ + k * KW_ + nt * 16, KW_);
      acc[nt] = wmma4(a, br, acc[nt]);
      acc[4 + nt] = wmma4(a, bi, acc[4 + nt]);
    }
  }
  float* Or = Yr + ch * (H_ * KW_) + wave * 16 * KW_;
  float* Oi = Yi + ch * (H_ * KW_) + wave * 16 * KW_;
#pragma unroll
  for (int nt = 0; nt < 4; ++nt) {
    store_c(Or + nt * 16, KW_, acc[nt]);
    store_c(Oi + nt * 16, KW_, acc[4 + nt]);
  }
}

// ---------------------------------------------------------------------------
// Stage 2 / Stage 4: complex H-axis transform per channel.
//   outR = HC@inR + S1@inI ;  outI = HC@inI + S2@inR
//   forward: S1=+sin, S2=-sin ;  inverse: S1=-sin, S2=+sin.
//   Data planes (2 x 128x64 = 64 KB) DMA'd into LDS; twiddles stream from L2.
// ---------------------------------------------------------------------------
__global__ __launch_bounds__(256)
void stage_hdft(const float* __restrict__ inR, const float* __restrict__ inI,
                float* __restrict__ outR, float* __restrict__ outI,
                const float* __restrict__ HC, const float* __restrict__ S1,
                const float* __restrict__ S2) {
  __shared__ alignas(16) float sY[2 * H_ * KW_];  // 65536 bytes
  long ch = blockIdx.x;
  const float* ir = inR + ch * (H_ * KW_);
  const float* ii = inI + ch * (H_ * KW_);
  async_copy_b128(sY, ir, H_ * KW_);
  async_copy_b128(sY + H_ * KW_, ii, H_ * KW_);
  async_join();
  int wave = threadIdx.x >> 5;
  v8f accR[4], accI[4];
#pragma unroll
  for (int n = 0; n < 4; ++n) { accR[n] = vzero8(); accI[n] = vzero8(); }
  const int arow = wave * 16 * H_;
  for (int k = 0; k < H_; k += 4) {
    v2f ac = load_a(HC + arow + k, H_);
    v2f a1 = load_a(S1 + arow + k, H_);
    v2f a2 = load_a(S2 + arow + k, H_);
#pragma unroll
    for (int nt = 0; nt < 4; ++nt) {
      v2f br = load_b(sY + k * KW_ + nt * 16, KW_);
      v2f bi = load_b(sY + H_ * KW_ + k * KW_ + nt * 16, KW_);
      accR[nt] = wmma4(ac, br, accR[nt]);
      accR[nt] = wmma4(a1, bi, accR[nt]);
      accI[nt] = wmma4(ac, bi, accI[nt]);
      accI[nt] = wmma4(a2, br, accI[nt]);
    }
  }
  float* orr = outR + ch * (H_ * KW_) + wave * 16 * KW_;
  float* oii = outI + ch * (H_ * KW_) + wave * 16 * KW_;
#pragma unroll
  for (int nt = 0; nt < 4; ++nt) {
    store_c(orr + nt * 16, KW_, accR[nt]);
    store_c(oii + nt * 16, KW_, accI[nt]);
  }
}

// ---------------------------------------------------------------------------
// Stage 3: spectral mixing at one frequency point per workgroup.
//   Out[b,o] = sum_i X[b,i] * W[i,o]   (complex 32x64x64 GEMM)
//   mode 0: W from pre-transposed dense layout [p][i][o] (w1/w2 blocks)
//   mode 1: W = 0.1 * sum_r la[e,r,i]*lb[e,o,r,p]   (LoRA expert, rank 4)
//   Launched 32 times in reference program order => exact overwrite semantics.
// ---------------------------------------------------------------------------
__global__ __launch_bounds__(256)
void mix_point(const float* __restrict__ Xr, const float* __restrict__ Xi,
               float* __restrict__ Or_, float* __restrict__ Oi_,
               const float* __restrict__ Wtr, const float* __restrict__ Wti,
               const float* __restrict__ la, const float* __restrict__ lb,
               int expert, int mode, int row0, int col0) {
  __shared__ alignas(16) float sWr[CIN_ * COUT_], sWi[CIN_ * COUT_];  // 32 KB
  __shared__ alignas(16) float sXr[B_ * CIN_], sXi[B_ * CIN_];        // 16 KB
  int px = blockIdx.x >> 4, py = blockIdx.x & 15;
  int kh = row0 + px, kw = col0 + py;
  int t = threadIdx.x;

  // gather the per-point X matrix [b][i] straight into LDS (async B32 DMA)
  for (int n = t; n < B_ * CIN_; n += 256) {
    unsigned off = (unsigned)(((n * H_ + kh) * KW_ + kw) * 4);  // n = b*CIN+i
    async_ld_b32(sXr + n, Xr, off);
    async_ld_b32(sXi + n, Xi, off);
  }

  if (mode == 0) {
    const float* sr = Wtr + (long)(px * 16 + py) * (CIN_ * COUT_);
    const float* si = Wti + (long)(px * 16 + py) * (CIN_ * COUT_);
    async_copy_b128(sWr, sr, CIN_ * COUT_);
    async_copy_b128(sWi, si, CIN_ * COUT_);
  } else {
    int p = px * 16 + py;
    for (int n = t; n < CIN_ * COUT_; n += 256) {
      int i = n >> 6, o = n & 63;
      float ar = 0.0f, ai = 0.0f;
#pragma unroll
      for (int r = 0; r < RANK_; ++r) {
        long lao = ((long)(expert * RANK_ + r) * CIN_ + i) * 2;
        float lar = la[lao], lai = la[lao + 1];
        long lbo = (((long)(expert * COUT_ + o) * RANK_ + r) * 256 + p) * 2;
        float lbr = lb[lbo], lbi = lb[lbo + 1];
        ar += lar * lbr - lai * lbi;
        ai += lar * lbi + lai * lbr;
      }
      sWr[n] = 0.1f * ar;
      sWi[n] = 0.1f * ai;
    }
  }
  async_join();

  int wave = t >> 5;
  int mt = wave & 1, nt = wave >> 1;  // M-tile (b), N-tile (o)
  v8f accR = vzero8(), accI = vzero8();
  const float* Ar = sXr + mt * 16 * CIN_;
  const float* Ai = sXi + mt * 16 * CIN_;
  const float* Br = sWr + nt * 16;
  const float* Bi = sWi + nt * 16;
  for (int k = 0; k < CIN_; k += 4) {
    v2f ar = load_a(Ar + k, CIN_);
    v2f ai = load_a(Ai + k, CIN_);
    v2f ain; ain.x = -ai.x; ain.y = -ai.y;  // fp32 WMMA has no A-neg modifier
    v2f br = load_b(Br + k * COUT_, COUT_);
    v2f bi = load_b(Bi + k * COUT_, COUT_);
    accR = wmma4(ar, br, accR);
    accR = wmma4(ain, bi, accR);   // Or = Xr@Wr - Xi@Wi
    accI = wmma4(ar, bi, accI);
    accI = wmma4(ai, br, accI);    // Oi = Xr@Wi + Xi@Wr
  }
  int lane = t & 31, col = lane & 15, half = lane >> 4;
  int o = nt * 16 + col;
#pragma unroll
  for (int j = 0; j < 8; ++j) {
    int b = mt * 16 + j + 8 * half;
    long off = ((long)(b * COUT_ + o) * H_ + kh) * KW_ + kw;
    Or_[off] = accR[j];
    Oi_[off] = accI[j];
  }
}

// ---------------------------------------------------------------------------
// Stage 5: Hermitian inverse W transform (real output), per channel:
//   Out[h][w] = Zr @ WR + Zi @ WI     (M=128, K=64, N=128)
// ---------------------------------------------------------------------------
__global__ __launch_bounds__(256)
void stage_iwdft(const float* __restrict__ Zr, const float* __restrict__ Zi,
                 const float* __restrict__ WR, const float* __restrict__ WI,
                 float* __restrict__ out) {
  __shared__ alignas(16) float sZ[2 * H_ * KW_];  // 65536 bytes
  long ch = blockIdx.x;
  const float* zr = Zr + ch * (H_ * KW_);
  const float* zi = Zi + ch * (H_ * KW_);
  async_copy_b128(sZ, zr, H_ * KW_);
  async_copy_b128(sZ + H_ * KW_, zi, H_ * KW_);
  async_join();
  int wave = threadIdx.x >> 5;
  v8f acc[8];
#pragma unroll
  for (int n = 0; n < 8; ++n) acc[n] = vzero8();
  const float* Ar = sZ + wave * 16 * KW_;
  const float* Ai = sZ + H_ * KW_ + wave * 16 * KW_;
  for (int k = 0; k < KW_; k += 4) {
    v2f ar = load_a(Ar + k, KW_);
    v2f ai = load_a(Ai + k, KW_);
#pragma unroll
    for (int nt = 0; nt < 8; ++nt) {
      v2f br = load_b(WR + k * W_ + nt * 16, W_);
      v2f bi = load_b(WI + k * W_ + nt * 16, W_);
      acc[nt] = wmma4(ar, br, acc[nt]);
      acc[nt] = wmma4(ai, bi, acc[nt]);
    }
  }
  float* O = out + ch * (H_ * W_) + wave * 16 * W_;
#pragma unroll
  for (int nt = 0; nt < 8; ++nt) store_c(O + nt * 16, W_, acc[nt]);
}

// ---------------------------------------------------------------------------
extern "C" void kernel_launch(void* const* d_in, const int* in_sizes, int n_in,
                              void* d_out, int out_size, void* d_ws,
                              size_t ws_size, hipStream_t stream) {
  (void)in_sizes; (void)n_in; (void)out_size; (void)ws_size;
  const float* x   = (const float*)d_in[0];
  const float* w1  = (const float*)d_in[1];  // complex interleaved
  const float* w2  = (const float*)d_in[2];
  const float* la1 = (const float*)d_in[3];
  const float* lb1 = (const float*)d_in[4];
  const float* la2 = (const float*)d_in[5];
  const float* lb2 = (const float*)d_in[6];
  float* out = (float*)d_out;
  float* ws = (float*)d_ws;

  const long PLANE = (long)B_ * CIN_ * H_ * KW_;  // 16,777,216 floats

  // workspace carve-up (all offsets 16B-aligned)
  float* CW  = ws;                 // 128*64
  float* SW  = CW + H_ * KW_;
  float* HC  = SW + H_ * KW_;      // 128*128
  float* HSP = HC + H_ * H_;
  float* HSN = HSP + H_ * H_;
  float* WR  = HSN + H_ * H_;      // 64*128
  float* WI  = WR + KW_ * W_;
  float* W1tr = WI + KW_ * W_;     // 256*64*64 each
  float* W1ti = W1tr + 256 * CIN_ * COUT_;
  float* W2tr = W1ti + 256 * CIN_ * COUT_;
  float* W2ti = W2tr + 256 * CIN_ * COUT_;
  float* Ar = W2ti + 256 * CIN_ * COUT_;  // buffer A: Y planes -> out_ft planes
  float* Ai = Ar + PLANE;
  float* Br = Ai + PLANE;                 // buffer B: x_ft planes -> z planes
  float* Bi = Br + PLANE;

  const int NCH = B_ * CIN_;  // 2048 (== B_*COUT_)

  build_tables<<<(H_ * H_ + 255) / 256, 256, 0, stream>>>(CW, SW, HC, HSP, HSN,
                                                          WR, WI);
  int nw = CIN_ * COUT_ * 256;
  transpose_w<<<(nw + 255) / 256, 256, 0, stream>>>(w1, W1tr, W1ti);
  transpose_w<<<(nw + 255) / 256, 256, 0, stream>>>(w2, W2tr, W2ti);

  // forward: W-axis real DFT, then H-axis complex DFT
  stage_wdft<<<NCH, 256, 0, stream>>>(x, CW, SW, Ar, Ai);
  stage_hdft<<<NCH, 256, 0, stream>>>(Ar, Ai, Br, Bi, HC, HSP, HSN);

  // out_ft := 0 (reuse buffer A; Br/Bi now hold x_ft)
  zero_buf<<<4096, 256, 0, stream>>>(Ar, 2 * PLANE);

  // spectral mixing in exact reference program order (same-stream serialization
  // reproduces the sequential .set overwrite semantics)
  mix_point<<<256, 256, 0, stream>>>(Br, Bi, Ar, Ai, W1tr, W1ti, nullptr,
                                     nullptr, 0, 0, 0, 0);
  mix_point<<<256, 256, 0, stream>>>(Br, Bi, Ar, Ai, W2tr, W2ti, nullptr,
                                     nullptr, 0, 0, H_ - M1_, 0);
  for (int i = 0; i < E_; ++i) {
    int r = (i + 1) / 4, c = (i + 1) % 4;
    mix_point<<<256, 256, 0, stream>>>(Br, Bi, Ar, Ai, nullptr, nullptr, la1,
                                       lb1, i, 1, M1_ * r, M2_ * c);
    // bound - re = 65 - 16*(r+1) = 49 - 16*r  (faithful to reference indexing)
    mix_point<<<256, 256, 0, stream>>>(Br, Bi, Ar, Ai, nullptr, nullptr, la2,
                                       lb2, i, 1, 49 - M1_ * r, M2_ * c);
  }

  // inverse: H-axis complex inverse DFT, then Hermitian irfft along W
  stage_hdft<<<NCH, 256, 0, stream>>>(Ar, Ai, Br, Bi, HC, HSN, HSP);
  stage_iwdft<<<NCH, 256, 0, stream>>>(Br, Bi, WR, WI, out);
}